// EquivariantCorrectionHead_88175678587465
// MI455X (gfx1250) — compile-verified
//
#include <hip/hip_runtime.h>
#include <cmath>

// ---------------------------------------------------------------------------
// EquivariantCorrectionHead on MI455X (gfx1250), fp32 WMMA (16x16x4) path.
// B=2048 batch, N_S=128 scalars, N_L2=8 l=2 channels, H=256 hidden.
// B-fragment reuse: 2 m-tiles per wave (y0, gemm) / 5 shared accs (y2, e).
// ---------------------------------------------------------------------------

typedef float v2f __attribute__((ext_vector_type(2)));
typedef float v8f __attribute__((ext_vector_type(8)));

#define B_  2048
#define NS_ 128
#define H_  256

__device__ __forceinline__ v8f wmma_f32_k4(v2f a, v2f b, v8f c) {
  // D = A(16x4 f32) x B(4x16 f32) + C(16x16 f32)
  return __builtin_amdgcn_wmma_f32_16x16x4_f32(false, a, false, b, (short)0, c,
                                               false, false);
}

// ---------------------------------------------------------------------------
// Wigner 3j (2e,2e)->2e, real basis, Frobenius norm 1.  125 floats -> ws.
// ---------------------------------------------------------------------------
__global__ void c222_init(float* __restrict__ C) {
  float Q[5][3][3];
  for (int a = 0; a < 5; ++a)
    for (int i = 0; i < 3; ++i)
      for (int j = 0; j < 3; ++j) Q[a][i][j] = 0.f;
  const float s  = 0.70710678118654752f;   // 1/sqrt(2)
  const float r6 = 0.40824829046386302f;   // 1/sqrt(6)
  Q[0][0][1] = s;  Q[0][1][0] = s;
  Q[1][1][2] = s;  Q[1][2][1] = s;
  Q[2][0][0] = -r6; Q[2][1][1] = -r6; Q[2][2][2] = 2.f * r6;
  Q[3][0][2] = s;  Q[3][2][0] = s;
  Q[4][0][0] = s;  Q[4][1][1] = -s;
  float Ct[125];
  float nrm = 0.f;
  for (int a = 0; a < 5; ++a)
    for (int b = 0; b < 5; ++b)
      for (int c = 0; c < 5; ++c) {
        float acc = 0.f;
        for (int i = 0; i < 3; ++i)
          for (int j = 0; j < 3; ++j)
            for (int k = 0; k < 3; ++k)
              acc += Q[a][i][j] * Q[b][j][k] * Q[c][k][i];
        Ct[(a * 5 + b) * 5 + c] = acc;
        nrm += acc * acc;
      }
  float inv = rsqrtf(nrm);
  for (int n = 0; n < 125; ++n) C[n] = Ct[n] * inv;
}

// ---------------------------------------------------------------------------
// prep: t[z][8][5] (7 selected + sum), ttS[z][64] = (1/sqrt5)*t.t^T,
//       Pp[k][z][64] = sum_ij C[i,j,k] t[a,i] t[b,j]
// ---------------------------------------------------------------------------
__global__ void prep_kernel(const float* __restrict__ kt, const float* __restrict__ C222,
                            float* __restrict__ t, float* __restrict__ ttS,
                            float* __restrict__ Pp, float inv_s5) {
  __shared__ float Cs[125];
  int tid = threadIdx.x;
  if (tid < 125) Cs[tid] = C222[tid];
  __syncthreads();
  int z = blockIdx.x * blockDim.x + tid;
  const float* kz = kt + (size_t)z * 40;
  float tl[8][5];
#pragma unroll
  for (int i = 0; i < 5; ++i) {
    float sum = 0.f;
#pragma unroll
    for (int a = 0; a < 8; ++a) sum += kz[a * 5 + i];
#pragma unroll
    for (int a = 0; a < 7; ++a) tl[a][i] = kz[a * 5 + i];
    tl[7][i] = sum;
  }
  float* tz = t + (size_t)z * 40;
#pragma unroll
  for (int a = 0; a < 8; ++a)
#pragma unroll
    for (int i = 0; i < 5; ++i) tz[a * 5 + i] = tl[a][i];

  float* ttz = ttS + (size_t)z * 64;
  for (int a = 0; a < 8; ++a)
    for (int b = 0; b < 8; ++b) {
      float acc = 0.f;
#pragma unroll
      for (int i = 0; i < 5; ++i) acc += tl[a][i] * tl[b][i];
      ttz[a * 8 + b] = inv_s5 * acc;
    }
  for (int k = 0; k < 5; ++k) {
    float* Pz = Pp + ((size_t)k * B_ + z) * 64;
    for (int a = 0; a < 8; ++a)
      for (int b = 0; b < 8; ++b) {
        float acc = 0.f;
#pragma unroll
        for (int i = 0; i < 5; ++i)
#pragma unroll
          for (int j = 0; j < 5; ++j)
            acc += Cs[(i * 5 + j) * 5 + k] * tl[a][i] * tl[b][j];
        Pz[a * 8 + b] = acc;
      }
  }
}

// Wc[u][a*256+w] = W022[u,a,w] + W202[a,u,w]        (128 x 2048)
__global__ void wc_kernel(const float* __restrict__ W022, const float* __restrict__ W202,
                          float* __restrict__ Wc) {
  int idx = blockIdx.x * blockDim.x + threadIdx.x;   // < 128*2048
  int u = idx >> 11, rem = idx & 2047;
  int a = rem >> 8, w = rem & 255;
  Wc[idx] = W022[((u << 3) + a) * 256 + w] + W202[((a << 7) + u) * 256 + w];
}

// Vc[m][n] = V022[m,n] + V202[n,m]                  (256 x 256)
__global__ void vc_kernel(const float* __restrict__ V022, const float* __restrict__ V202,
                          float* __restrict__ Vc) {
  int idx = blockIdx.x * blockDim.x + threadIdx.x;   // < 65536
  int m = idx >> 8, n = idx & 255;
  Vc[idx] = V022[idx] + V202[n * 256 + m];
}

// ---------------------------------------------------------------------------
// y0[z][w] = C0 * ( sum_{u,v} s_u s_v W000[u,v,w]  +  tt_scaled . W220 )
// K=16384 WMMA GEMM with on-the-fly outer-product A-fragments.
// Each wave owns ONE 16-wide w tile but TWO 16-row z tiles -> every W000
// B-fragment feeds two v_wmma (halves the dominant L2 traffic).
// block = (32,4): 4 waves; grid = (4, B/32).
// ---------------------------------------------------------------------------
__global__ void y0_kernel(const float* __restrict__ s, const float* __restrict__ W000,
                          const float* __restrict__ ttS, const float* __restrict__ W220,
                          float* __restrict__ y0, float c0) {
  __shared__ float sT[32 * 128];
  int tid = threadIdx.y * 32 + threadIdx.x;          // 0..127
  int zbase = blockIdx.y * 32;
#pragma unroll
  for (int i = 0; i < 32; ++i)
    sT[tid + i * 128] = s[(size_t)zbase * 128 + tid + i * 128];
  __syncthreads();

  int lane = threadIdx.x, half = lane >> 4, l16 = lane & 15;
  int n0 = (blockIdx.x * 4 + threadIdx.y) * 16;
  const float* srow0 = &sT[l16 * 128];
  const float* srow1 = &sT[(16 + l16) * 128];
  v8f acc0 = {0.f, 0.f, 0.f, 0.f, 0.f, 0.f, 0.f, 0.f};
  v8f acc1 = acc0;

  for (int u = 0; u < 128; ++u) {
    float su0 = srow0[u];
    float su1 = srow1[u];
    const float* Bp = W000 + ((size_t)u * 128) * 256 + n0 + l16;
#pragma unroll 4
    for (int v0 = 0; v0 < 128; v0 += 4) {
      int kv = v0 + 2 * half;
      v2f b;  b.x  = Bp[(size_t)kv * 256];   b.y  = Bp[(size_t)kv * 256 + 256];
      v2f a0; a0.x = su0 * srow0[kv];        a0.y = su0 * srow0[kv + 1];
      v2f a1; a1.x = su1 * srow1[kv];        a1.y = su1 * srow1[kv + 1];
      acc0 = wmma_f32_k4(a0, b, acc0);
      acc1 = wmma_f32_k4(a1, b, acc1);
    }
  }
  // small K=64 tail: tt_scaled @ W220 (shared B-fragment, two tiles)
  const float* ttr0 = ttS + (size_t)(zbase + l16) * 64;
  const float* ttr1 = ttS + (size_t)(zbase + 16 + l16) * 64;
#pragma unroll
  for (int k0 = 0; k0 < 64; k0 += 4) {
    int ka = k0 + 2 * half;
    v2f b;  b.x  = W220[(size_t)ka * 256 + n0 + l16];
    b.y  = W220[(size_t)(ka + 1) * 256 + n0 + l16];
    v2f a0; a0.x = ttr0[ka]; a0.y = ttr0[ka + 1];
    v2f a1; a1.x = ttr1[ka]; a1.y = ttr1[ka + 1];
    acc0 = wmma_f32_k4(a0, b, acc0);
    acc1 = wmma_f32_k4(a1, b, acc1);
  }
#pragma unroll
  for (int r = 0; r < 8; ++r) {
    y0[(size_t)(zbase + r + 8 * half) * 256 + n0 + l16]      = c0 * acc0[r];
    y0[(size_t)(zbase + 16 + r + 8 * half) * 256 + n0 + l16] = c0 * acc1[r];
  }
}

// ---------------------------------------------------------------------------
// Generic fp32 WMMA GEMM, 2 m-tiles per wave:  C[M,N] = alpha * A[M,K] x B[K,N]
// block=(32,4): wave ty owns n-tile blockIdx.x*4+ty; grid=(N/64, M/32).
// ---------------------------------------------------------------------------
__global__ void gemm16x2_kernel(const float* __restrict__ A, const float* __restrict__ Bm,
                                float* __restrict__ C, int K, int lda, int ldb, int ldc,
                                float alpha) {
  int lane = threadIdx.x, half = lane >> 4, l16 = lane & 15;
  int n0 = (blockIdx.x * 4 + threadIdx.y) * 16;
  int m0 = blockIdx.y * 32;
  const float* Ar0 = A + (size_t)(m0 + l16) * lda;
  const float* Ar1 = A + (size_t)(m0 + 16 + l16) * lda;
  const float* Bc  = Bm + n0 + l16;
  v8f acc0 = {0.f, 0.f, 0.f, 0.f, 0.f, 0.f, 0.f, 0.f};
  v8f acc1 = acc0;
  for (int k0 = 0; k0 < K; k0 += 4) {
    int ka = k0 + 2 * half;
    v2f b;  b.x  = Bc[(size_t)ka * ldb]; b.y = Bc[(size_t)(ka + 1) * ldb];
    v2f a0; a0.x = Ar0[ka];              a0.y = Ar0[ka + 1];
    v2f a1; a1.x = Ar1[ka];              a1.y = Ar1[ka + 1];
    acc0 = wmma_f32_k4(a0, b, acc0);
    acc1 = wmma_f32_k4(a1, b, acc1);
  }
#pragma unroll
  for (int r = 0; r < 8; ++r) {
    C[(size_t)(m0 + r + 8 * half) * ldc + n0 + l16]      = alpha * acc0[r];
    C[(size_t)(m0 + 16 + r + 8 * half) * ldc + n0 + l16] = alpha * acc1[r];
  }
}

// ---------------------------------------------------------------------------
// y2 planes: y2p[k][z][w] = C2*( (P_k @ W222)[z,w] + inv_s5*sum_a Aw[z,a,w]*t[z,a,k] )
// W222 B-fragment shared across all 5 k accumulators.
// ---------------------------------------------------------------------------
__global__ void y2_kernel(const float* __restrict__ Pp, const float* __restrict__ W222,
                          const float* __restrict__ Aw, const float* __restrict__ t,
                          float* __restrict__ y2p, float c2, float inv_s5) {
  __shared__ float tT[16 * 40];
  int tid = threadIdx.y * 32 + threadIdx.x;
  int zbase = blockIdx.y * 16;
  for (int i = tid; i < 640; i += 128) tT[i] = t[(size_t)zbase * 40 + i];
  __syncthreads();

  int lane = threadIdx.x, half = lane >> 4, l16 = lane & 15;
  int n0 = (blockIdx.x * 4 + threadIdx.y) * 16;
  v8f zero = {0.f, 0.f, 0.f, 0.f, 0.f, 0.f, 0.f, 0.f};
  v8f acc[5];
#pragma unroll
  for (int k = 0; k < 5; ++k) acc[k] = zero;

#pragma unroll 4
  for (int k0 = 0; k0 < 64; k0 += 4) {
    int ka = k0 + 2 * half;
    v2f b; b.x = W222[(size_t)ka * 256 + n0 + l16];
    b.y = W222[(size_t)(ka + 1) * 256 + n0 + l16];
#pragma unroll
    for (int k = 0; k < 5; ++k) {
      const float* Pk = Pp + ((size_t)k * B_ + zbase + l16) * 64;
      v2f a; a.x = Pk[ka]; a.y = Pk[ka + 1];
      acc[k] = wmma_f32_k4(a, b, acc[k]);
    }
  }
#pragma unroll
  for (int r = 0; r < 8; ++r) {
    int zi = r + 8 * half;
    int z = zbase + zi;
    int w = n0 + l16;
    float avv[8];
#pragma unroll
    for (int a = 0; a < 8; ++a) avv[a] = Aw[(size_t)z * 2048 + a * 256 + w];
#pragma unroll
    for (int k = 0; k < 5; ++k) {
      float dot = 0.f;
#pragma unroll
      for (int a = 0; a < 8; ++a) dot += avv[a] * tT[zi * 40 + a * 5 + k];
      y2p[((size_t)k * B_ + z) * 256 + w] = c2 * (acc[k][r] + inv_s5 * dot);
    }
  }
}

// ---------------------------------------------------------------------------
// e planes fused: e[i][z][v] = (y2_i @ V222)[z,v], i = 0..4.
// Single kernel; every V222 B-fragment feeds 5 v_wmma (one per plane).
// ---------------------------------------------------------------------------
__global__ void e_kernel(const float* __restrict__ y2p, const float* __restrict__ V222,
                         float* __restrict__ e) {
  int lane = threadIdx.x, half = lane >> 4, l16 = lane & 15;
  int n0 = (blockIdx.x * 4 + threadIdx.y) * 16;
  int m0 = blockIdx.y * 16;
  v8f zero = {0.f, 0.f, 0.f, 0.f, 0.f, 0.f, 0.f, 0.f};
  v8f acc[5];
#pragma unroll
  for (int i = 0; i < 5; ++i) acc[i] = zero;
  const float* Bc = V222 + n0 + l16;
  for (int k0 = 0; k0 < 256; k0 += 4) {
    int ka = k0 + 2 * half;
    v2f b; b.x = Bc[(size_t)ka * 256]; b.y = Bc[(size_t)(ka + 1) * 256];
#pragma unroll
    for (int i = 0; i < 5; ++i) {
      const float* Ar = y2p + ((size_t)i * B_ + m0 + l16) * 256;
      v2f a; a.x = Ar[ka]; a.y = Ar[ka + 1];
      acc[i] = wmma_f32_k4(a, b, acc[i]);
    }
  }
#pragma unroll
  for (int i = 0; i < 5; ++i)
#pragma unroll
    for (int r = 0; r < 8; ++r)
      e[((size_t)i * B_ + m0 + r + 8 * half) * 256 + n0 + l16] = acc[i][r];
}

// ---------------------------------------------------------------------------
// out[z][k] = C2B*( inv_s5*sum_v a[z,v]*y2[z,v,k] + sum_{v,i,j} e_i[z,v] C[i,j,k] y2[z,v,j] )
// one block per z, thread v in 0..255, LDS tree reduction.
// ---------------------------------------------------------------------------
__global__ void final_kernel(const float* __restrict__ av, const float* __restrict__ y2p,
                             const float* __restrict__ ev, const float* __restrict__ C222,
                             float* __restrict__ out, float c2b, float inv_s5) {
  __shared__ float Cs[125];
  __shared__ float red[5][256];
  int v = threadIdx.x;
  if (v < 125) Cs[v] = C222[v];
  __syncthreads();
  int z = blockIdx.x;
  float y2v[5], e_[5];
#pragma unroll
  for (int j = 0; j < 5; ++j) y2v[j] = y2p[((size_t)j * B_ + z) * 256 + v];
#pragma unroll
  for (int i = 0; i < 5; ++i) e_[i] = ev[((size_t)i * B_ + z) * 256 + v];
  float a0 = av[(size_t)z * 256 + v];
  float pk[5];
#pragma unroll
  for (int k = 0; k < 5; ++k) pk[k] = inv_s5 * a0 * y2v[k];
#pragma unroll
  for (int i = 0; i < 5; ++i)
#pragma unroll
    for (int j = 0; j < 5; ++j) {
      float prod = e_[i] * y2v[j];
#pragma unroll
      for (int k = 0; k < 5; ++k) pk[k] += Cs[(i * 5 + j) * 5 + k] * prod;
    }
#pragma unroll
  for (int k = 0; k < 5; ++k) red[k][v] = pk[k];
  __syncthreads();
  for (int stp = 128; stp > 0; stp >>= 1) {
    if (v < stp) {
#pragma unroll
      for (int k = 0; k < 5; ++k) red[k][v] += red[k][v + stp];
    }
    __syncthreads();
  }
  if (v < 5) out[(size_t)z * 5 + v] = c2b * red[v][0];
}

// ---------------------------------------------------------------------------
extern "C" void kernel_launch(void* const* d_in, const int* in_sizes, int n_in,
                              void* d_out, int out_size, void* d_ws, size_t ws_size,
                              hipStream_t stream) {
  const float* scalars = (const float*)d_in[0];
  const float* kt      = (const float*)d_in[1];
  const float* W000    = (const float*)d_in[2];
  const float* W220    = (const float*)d_in[3];
  const float* W022    = (const float*)d_in[4];
  const float* W202    = (const float*)d_in[5];
  const float* W222    = (const float*)d_in[6];
  const float* V022    = (const float*)d_in[7];
  const float* V202    = (const float*)d_in[8];
  const float* V222    = (const float*)d_in[9];
  float* out = (float*)d_out;
  float* ws  = (float*)d_ws;

  // workspace layout (floats), total ~46.7 MB
  float* C222d = ws;                                   // 128
  float* t     = C222d + 128;                          // 2048*40
  float* ttS   = t     + (size_t)B_ * 40;              // 2048*64
  float* Pp    = ttS   + (size_t)B_ * 64;              // 5*2048*64
  float* Wc    = Pp    + (size_t)5 * B_ * 64;          // 128*2048
  float* Vc    = Wc    + (size_t)128 * 2048;           // 256*256
  float* y0    = Vc    + (size_t)256 * 256;            // 2048*256
  float* Aw    = y0    + (size_t)B_ * 256;             // 2048*2048
  float* y2p   = Aw    + (size_t)B_ * 2048;            // 5*2048*256
  float* acmb  = y2p   + (size_t)5 * B_ * 256;         // 2048*256
  float* e     = acmb  + (size_t)B_ * 256;             // 5*2048*256

  const float C0f  = (float)(1.0 / sqrt((double)(128 * 128 + 8 * 8)));
  const float C2f  = (float)sqrt(5.0 / (2.0 * 128 * 8 + 64.0));
  const float C2Bf = (float)sqrt(5.0 / (3.0 * 256.0 * 256.0));
  const float IS5  = (float)(1.0 / sqrt(5.0));

  dim3 wblk(32, 4);  // 4 wave32's

  c222_init<<<dim3(1), dim3(1), 0, stream>>>(C222d);
  prep_kernel<<<dim3(B_ / 128), dim3(128), 0, stream>>>(kt, C222d, t, ttS, Pp, IS5);
  wc_kernel<<<dim3((128 * 2048) / 256), dim3(256), 0, stream>>>(W022, W202, Wc);
  vc_kernel<<<dim3((256 * 256) / 256), dim3(256), 0, stream>>>(V022, V202, Vc);

  // y0 : big quadratic form + tt tail (dominant compute, fp32 WMMA, 2 z-tiles/wave)
  y0_kernel<<<dim3(4, B_ / 32), wblk, 0, stream>>>(scalars, W000, ttS, W220, y0, C0f);

  // Aw[z, a*256+w] = s @ Wc     (M=2048, K=128, N=2048)
  gemm16x2_kernel<<<dim3(32, B_ / 32), wblk, 0, stream>>>(scalars, Wc, Aw,
                                                          128, 128, 2048, 2048, 1.0f);

  // y2 planes
  y2_kernel<<<dim3(4, B_ / 16), wblk, 0, stream>>>(Pp, W222, Aw, t, y2p, C2f, IS5);

  // a = y0 @ Vc                 (M=2048, K=256, N=256)
  gemm16x2_kernel<<<dim3(4, B_ / 32), wblk, 0, stream>>>(y0, Vc, acmb,
                                                         256, 256, 256, 256, 1.0f);
  // e_i = y2_i @ V222, i=0..4   (fused, shared B-fragments)
  e_kernel<<<dim3(4, B_ / 16), wblk, 0, stream>>>(y2p, V222, e);

  final_kernel<<<dim3(B_), dim3(256), 0, stream>>>(acmb, y2p, e, C222d, out, C2Bf, IS5);

  (void)in_sizes; (void)n_in; (void)out_size; (void)ws_size;
}